// MinibatchDiscrimination_15985868275802
// MI455X (gfx1250) — compile-verified
//
#include <hip/hip_runtime.h>

// Problem constants (from the reference)
#define B_   512
#define IN_  512
#define OUT_ 64
#define KD_  16
#define MW   1024   // OUT_*KD_ : width of the M matrix
#define OW   576    // IN_+OUT_ : width of the output

typedef __attribute__((ext_vector_type(2))) float v2f;
typedef __attribute__((ext_vector_type(8))) float v8f;
// Exact pointee type the async-to-LDS builtin expects (per hipcc diagnostic):
typedef int v4i_raw __attribute__((vector_size(16)));
typedef __attribute__((address_space(1))) v4i_raw* gptr_b128;
typedef __attribute__((address_space(3))) v4i_raw* lptr_b128;

// gfx1250 async global->LDS staging (ASYNCcnt-tracked), if the toolchain has it
#if defined(__has_builtin)
#if __has_builtin(__builtin_amdgcn_global_load_async_to_lds_b128) && \
    __has_builtin(__builtin_amdgcn_s_wait_asynccnt)
#define USE_ASYNC_LDS 1
#endif
#endif

// -----------------------------------------------------------------------------
// Stage 1: M = x @ reshape(T, [IN, OUT*KD]) via V_WMMA_F32_16X16X4_F32.
// One wave32 per 16x64 output tile: the A-fragment (b64 per lane) is reused
// across 4 WMMAs -> 2.25 vmem per wmma instead of 3, and 4 independent
// accumulator chains for the WMMA RAW-hazard spacing.
//
// A (16x4 f32) lane layout (ISA 7.12.2):
//   lanes 0-15 : VGPR0 = A[row][K=0], VGPR1 = A[row][K=1]
//   lanes16-31 : VGPR0 = A[row][K=2], VGPR1 = A[row][K=3]
// => per lane one contiguous float2 at column k + 2*(lane>>4).
// B (4x16) mirrors it (K rows strided by MW). C/D: lane>=16 -> row+8.
// -----------------------------------------------------------------------------
__global__ __launch_bounds__(32)
void mbdisc_gemm_wmma(const float* __restrict__ x,
                      const float* __restrict__ T,
                      float* __restrict__ M) {
    const int tileM = blockIdx.x;          // 0..31  (16-row bands of x / M)
    const int tileN = blockIdx.y;          // 0..15  (64-col bands of M)
    const int lane  = threadIdx.x & 31;
    const int half  = lane >> 4;           // 0 or 1
    const int l15   = lane & 15;

    const int rowA  = tileM * 16 + l15;    // A row this lane supplies
    const int colB0 = tileN * 64 + l15;    // first B col this lane supplies

    const float* __restrict__ xrow = x + rowA * IN_ + 2 * half;
    const float* __restrict__ tcol = T + 2 * half * MW + colB0;

    v8f acc0 = {}, acc1 = {}, acc2 = {}, acc3 = {};
    for (int k = 0; k < IN_; k += 4) {
        v2f a = *(const v2f*)(xrow + k);           // x[rowA][k+2h .. k+2h+1]
        const float* __restrict__ t0 = tcol + k * MW;
        v2f b0, b1, b2, b3;
        b0.x = t0[0];       b0.y = t0[MW +  0];    // T[k+2h][col], T[k+2h+1][col]
        b1.x = t0[16];      b1.y = t0[MW + 16];
        b2.x = t0[32];      b2.y = t0[MW + 32];
        b3.x = t0[48];      b3.y = t0[MW + 48];
        acc0 = __builtin_amdgcn_wmma_f32_16x16x4_f32(false, a, false, b0, (short)0, acc0, false, false);
        acc1 = __builtin_amdgcn_wmma_f32_16x16x4_f32(false, a, false, b1, (short)0, acc1, false, false);
        acc2 = __builtin_amdgcn_wmma_f32_16x16x4_f32(false, a, false, b2, (short)0, acc2, false, false);
        acc3 = __builtin_amdgcn_wmma_f32_16x16x4_f32(false, a, false, b3, (short)0, acc3, false, false);
    }

    // Store D tiles: vgpr r, lane<16 -> row r, lane>=16 -> row r+8
    const int outRow = tileM * 16 + half * 8;
    float* __restrict__ Mp = M + outRow * MW + colB0;
#pragma unroll
    for (int r = 0; r < 8; ++r) {
        Mp[r * MW +  0] = acc0[r];
        Mp[r * MW + 16] = acc1[r];
        Mp[r * MW + 32] = acc2[r];
        Mp[r * MW + 48] = acc3[r];
    }
}

// -----------------------------------------------------------------------------
// Stage 2: out[b, 512+o] = sum_{b2} exp(-sum_k |M[b,o,k]-M[b2,o,k]|) - 1
// Grid = (64 features) x (4 row-chunks of 128) for occupancy; each block
// stages the full M[:, o, :] slab (512x16 f32) into LDS.
// SSTRIDE=20 floats = 80 B: keeps each 16-float row 16B-aligned for async
// b128 writes and de-phases own-row reads across the 64 LDS banks.
// -----------------------------------------------------------------------------
#define SSTRIDE 20
__global__ __launch_bounds__(128)
void mbdisc_pairwise(const float* __restrict__ M,
                     float* __restrict__ out) {
    __shared__ float sM[B_ * SSTRIDE];     // 512*20*4 = 40 KB (<< 320 KB/WGP)
    const int o   = blockIdx.x;            // 0..63
    const int b   = blockIdx.y * 128 + threadIdx.x;   // this thread's row
    const int tid = threadIdx.x;

#ifdef USE_ASYNC_LDS
    // Async DMA-style staging: per row, four 16-byte async copies to LDS.
    for (int r = tid; r < B_; r += 128) {
        const float* __restrict__ g = M + r * MW + o * KD_;
        float* l = &sM[r * SSTRIDE];
#pragma unroll
        for (int q = 0; q < 4; ++q) {
            __builtin_amdgcn_global_load_async_to_lds_b128(
                (gptr_b128)(g + q * 4),
                (lptr_b128)(l + q * 4),
                0, 0);
        }
    }
    __builtin_amdgcn_s_wait_asynccnt(0);
#else
    for (int idx = tid; idx < B_ * KD_; idx += 128) {
        const int b2 = idx >> 4;
        const int k  = idx & 15;
        sM[b2 * SSTRIDE + k] = M[b2 * MW + o * KD_ + k];
    }
#endif
    __syncthreads();

    float m[KD_];
#pragma unroll
    for (int k = 0; k < KD_; ++k) m[k] = sM[b * SSTRIDE + k];

    float acc = 0.0f;
    for (int b2 = 0; b2 < B_; ++b2) {
        const float* __restrict__ row = sM + b2 * SSTRIDE;  // broadcast reads
        float l1 = 0.0f;
#pragma unroll
        for (int k = 0; k < KD_; ++k)
            l1 += fabsf(m[k] - row[k]);     // v_sub + v_add with |src| modifier
        acc += __expf(-l1);                 // v_exp_f32
    }
    out[b * OW + IN_ + o] = acc - 1.0f;
}

// -----------------------------------------------------------------------------
// Stage 3: out[:, :512] = x
// -----------------------------------------------------------------------------
__global__ __launch_bounds__(256)
void mbdisc_copy_x(const float* __restrict__ x, float* __restrict__ out) {
    const int idx = blockIdx.x * 256 + threadIdx.x;   // over B_*IN_ = 262144
    const int b = idx >> 9;                           // /512
    const int i = idx & (IN_ - 1);
    out[b * OW + i] = x[idx];
}

extern "C" void kernel_launch(void* const* d_in, const int* in_sizes, int n_in,
                              void* d_out, int out_size, void* d_ws, size_t ws_size,
                              hipStream_t stream) {
    const float* x = (const float*)d_in[0];   // [512, 512]
    const float* T = (const float*)d_in[1];   // [512, 64, 16]
    float* out = (float*)d_out;               // [512, 576]
    float* M   = (float*)d_ws;                // [512, 1024] scratch (2 MB)

    // Stage 1: GEMM via WMMA f32 16x16x4, 16x64 tile per wave
    dim3 ggrid(B_ / 16, MW / 64);             // 32 x 16 tiles
    mbdisc_gemm_wmma<<<ggrid, 32, 0, stream>>>(x, T, M);

    // Stage 3 (independent of M): copy x into out[:, :512]
    mbdisc_copy_x<<<(B_ * IN_) / 256, 256, 0, stream>>>(x, out);

    // Stage 2: pairwise exp(-L1) reduction, 64 features x 4 row-chunks
    dim3 pgrid(OUT_, B_ / 128);               // 64 x 4 blocks, 128 thr each
    mbdisc_pairwise<<<pgrid, 128, 0, stream>>>(M, out);
}